// GCN_33217277067304
// MI455X (gfx1250) — compile-verified
//
#include <hip/hip_runtime.h>
#include <math.h>

typedef __attribute__((ext_vector_type(2))) float v2f;
typedef __attribute__((ext_vector_type(8))) float v8f;

#define THREADS 256

// ---------------- degree computation (integer atomics -> rsqrt) --------------
__global__ void k_deg_init(int* __restrict__ deg, int n) {
  int i = blockIdx.x * blockDim.x + threadIdx.x;
  if (i < n) deg[i] = 1;  // self-loop
}

__global__ void k_deg_accum(int* __restrict__ deg, const int* __restrict__ dst, int e) {
  int i = blockIdx.x * blockDim.x + threadIdx.x;
  if (i < e) atomicAdd(&deg[dst[i]], 1);
}

__global__ void k_dinv(const int* __restrict__ deg, float* __restrict__ dinv, int n) {
  int i = blockIdx.x * blockDim.x + threadIdx.x;
  if (i < n) dinv[i] = rsqrtf((float)deg[i]);
}

// ---------------- f32 WMMA GEMM: C = A(MxK) * B(KxN), fused self-loop init ---
// One 16x16 tile per wave; blockIdx.x = M tile; wave id = N tile.
// Also writes Cself = C * dinv[m]^2 (initializes the scatter accumulator with
// the self-loop contribution, so no memset pass is needed).
template <int K, int N>
__global__ void k_gemm_wmma(const float* __restrict__ A, const float* __restrict__ B,
                            const float* __restrict__ dinv,
                            float* __restrict__ C, float* __restrict__ Cself) {
  const int lane   = threadIdx.x & 31;
  const int wave   = threadIdx.x >> 5;
  const int m_base = blockIdx.x << 4;
  const int n_base = wave << 4;
  const int r      = lane & 15;          // row of A / col of B handled by lane
  const int kk     = (lane >> 4) << 1;   // K sub-offset: 0 (lanes 0-15) or 2 (16-31)

  const float* Arow = A + (size_t)(m_base + r) * K;
  const float* Bp   = B + n_base + r;

  v8f acc = {};
#pragma unroll 8
  for (int k = 0; k < K; k += 4) {
    v2f a = *(const v2f*)(Arow + k + kk);       // A[m][k+kk], A[m][k+kk+1]
    v2f b;
    b.x = Bp[(size_t)(k + kk) * N];             // B[k+kk][n]
    b.y = Bp[(size_t)(k + kk + 1) * N];         // B[k+kk+1][n]
    acc = __builtin_amdgcn_wmma_f32_16x16x4_f32(
        /*neg_a=*/false, a, /*neg_b=*/false, b,
        /*c_mod=*/(short)0, acc, /*reuse_a=*/false, /*reuse_b=*/false);
  }

  const int n  = n_base + r;
  const int m0 = m_base + ((lane >> 4) << 3);   // C/D layout: lanes 16-31 -> M+8
#pragma unroll
  for (int v = 0; v < 8; ++v) {
    int   m   = m0 + v;
    float val = acc[v];
    C[(size_t)m * N + n] = val;
    float di = dinv[m];
    Cself[(size_t)m * N + n] = val * di * di;   // self-loop message
  }
}

// ---------------- edge scatter: agg[dst] += xw[src] * dinv[src]*dinv[dst] ----
template <int F>
__global__ void k_scatter(const float* __restrict__ xw, const float* __restrict__ dinv,
                          const int* __restrict__ src, const int* __restrict__ dst,
                          float* __restrict__ agg, int e) {
  constexpr int TPE = F >> 2;  // threads per edge (4 floats per thread)
  int t = blockIdx.x * blockDim.x + threadIdx.x;
  if (t >= e * TPE) return;
  int ed = t / TPE;                 // power of 2 -> shift
  int f0 = (t - ed * TPE) << 2;
  int s = src[ed], d = dst[ed];
  float nrm = dinv[s] * dinv[d];
  const float4 v = *(const float4*)(xw + (size_t)s * F + f0);
  float* po = agg + (size_t)d * F + f0;
  unsafeAtomicAdd(po + 0, v.x * nrm);   // global_atomic_add_f32 (no return)
  unsafeAtomicAdd(po + 1, v.y * nrm);
  unsafeAtomicAdd(po + 2, v.z * nrm);
  unsafeAtomicAdd(po + 3, v.w * nrm);
}

// ---------------- bias + ReLU (in place) -------------------------------------
template <int F>
__global__ void k_bias_relu(float* __restrict__ h, const float* __restrict__ b, int total) {
  int t = blockIdx.x * blockDim.x + threadIdx.x;
  if (t < total) h[t] = fmaxf(h[t] + b[t & (F - 1)], 0.0f);
}

// ---------------- bias + log_softmax over 64 cols: one wave32 per row --------
__global__ void k_bias_logsoftmax64(const float* __restrict__ in, const float* __restrict__ b,
                                    float* __restrict__ out, int nrows) {
  int row = blockIdx.x * (blockDim.x >> 5) + (threadIdx.x >> 5);
  if (row >= nrows) return;
  int lane = threadIdx.x & 31;
  const float* pi = in + (size_t)row * 64;
  float v0 = pi[lane]      + b[lane];
  float v1 = pi[lane + 32] + b[lane + 32];
  float m = fmaxf(v0, v1);
#pragma unroll
  for (int off = 16; off; off >>= 1) m = fmaxf(m, __shfl_xor(m, off, 32));
  float s = expf(v0 - m) + expf(v1 - m);
#pragma unroll
  for (int off = 16; off; off >>= 1) s += __shfl_xor(s, off, 32);
  float ls = m + logf(s);
  float* po = out + (size_t)row * 64;
  po[lane]      = v0 - ls;
  po[lane + 32] = v1 - ls;
}

// ---------------------------------------------------------------------------
extern "C" void kernel_launch(void* const* d_in, const int* in_sizes, int n_in,
                              void* d_out, int out_size, void* d_ws, size_t ws_size,
                              hipStream_t stream) {
  const float* x  = (const float*)d_in[0];
  const int*   ei = (const int*)d_in[1];
  const float* W1 = (const float*)d_in[2];
  const float* b1 = (const float*)d_in[3];
  const float* W2 = (const float*)d_in[4];
  const float* b2 = (const float*)d_in[5];
  float* out = (float*)d_out;

  const int Nn = in_sizes[0] / 128;  // 50000 (multiple of 16)
  const int E  = in_sizes[1] / 2;    // 800000
  const int* src = ei;
  const int* dst = ei + E;

  // workspace layout (256-element padded)
  const size_t pad = (size_t)((Nn + 255) & ~255);
  int*   deg  = (int*)d_ws;
  float* dinv = (float*)d_ws + pad;
  float* xw   = dinv + pad;                 // N x 128
  float* agg1 = xw   + (size_t)Nn * 128;    // N x 128 (becomes h after ReLU)
  float* hw   = agg1 + (size_t)Nn * 128;    // N x 64
  float* agg2 = hw   + (size_t)Nn * 64;     // N x 64

  const int gN = (Nn + THREADS - 1) / THREADS;
  const int gE = (E + THREADS - 1) / THREADS;

  // degrees -> dinv
  k_deg_init<<<gN, THREADS, 0, stream>>>(deg, Nn);
  k_deg_accum<<<gE, THREADS, 0, stream>>>(deg, dst, E);
  k_dinv<<<gN, THREADS, 0, stream>>>(deg, dinv, Nn);

  // layer 1: xw = x @ W1 ; agg1 initialized with self-loop term
  k_gemm_wmma<128, 128><<<Nn / 16, 8 * 32, 0, stream>>>(x, W1, dinv, xw, agg1);
  {
    int total = E * 32;  // F=128 -> 32 threads/edge
    k_scatter<128><<<(total + THREADS - 1) / THREADS, THREADS, 0, stream>>>(
        xw, dinv, src, dst, agg1, E);
  }
  k_bias_relu<128><<<(Nn * 128 + THREADS - 1) / THREADS, THREADS, 0, stream>>>(
      agg1, b1, Nn * 128);

  // layer 2: hw = h @ W2 ; agg2 initialized with self-loop term
  k_gemm_wmma<128, 64><<<Nn / 16, 4 * 32, 0, stream>>>(agg1, W2, dinv, hw, agg2);
  {
    int total = E * 16;  // F=64 -> 16 threads/edge
    k_scatter<64><<<(total + THREADS - 1) / THREADS, THREADS, 0, stream>>>(
        hw, dinv, src, dst, agg2, E);
  }

  // bias + log_softmax -> out
  k_bias_logsoftmax64<<<(Nn + 7) / 8, THREADS, 0, stream>>>(agg2, b2, out, Nn);
}